// Retraction_48679159333527
// MI455X (gfx1250) — compile-verified
//
#include <hip/hip_runtime.h>
#include <math.h>

#define N_BATCH 256
#define P_DIM   1024
#define Q_DIM   128
#define EPSV    1e-6f
#define NS_ITERS 16
#define PITCH   144   // pair-row pitch: 2*144 % 64banks == 32 -> halves on disjoint banks

typedef __attribute__((ext_vector_type(2))) float v2f;
typedef __attribute__((ext_vector_type(8))) float v8f;

__device__ __forceinline__ v8f wmma_f32(v2f a, v2f b, v8f c) {
    return __builtin_amdgcn_wmma_f32_16x16x4_f32(
        false, a, false, b, (short)0, c, false, false);
}

// element (k,q) lives at ((k>>1)*PITCH + q)*2 + (k&1); kk is even -> single b64
__device__ __forceinline__ v2f pairfrag(const float* lds, int kk, int q) {
    return *(const v2f*)&lds[((kk >> 1) * PITCH + q) * 2];
}

// ---------------------------------------------------------------------------
// Kernel 1: gram[n] = temp^T @ temp, temp = X - lr*G (1024x128 -> 128x128).
// One block/batch, 8 waves; wave = (pair-of-row-blocks, half-of-col-blocks).
// temp chunk staged K-pair-interleaved so A and B frags are single b64 loads.
// ---------------------------------------------------------------------------
__global__ __launch_bounds__(256) void gram_kernel(const float* __restrict__ X,
                                                   const float* __restrict__ G,
                                                   const float* __restrict__ lr_p,
                                                   float* __restrict__ gram) {
    __shared__ __attribute__((aligned(16))) float lds[32 * PITCH * 2]; // 64 k-rows
    const int n    = blockIdx.x;
    const float lr = lr_p[0];
    const int tid  = threadIdx.x;
    const int wave = tid >> 5;
    const int lane = tid & 31;
    const int half = lane >> 4;
    const int lid  = lane & 15;
    const int m0   = (wave & 3) * 32;   // two 16-row blocks
    const int n0   = (wave >> 2) * 64;  // four 16-col blocks

    const float* Xb = X + (size_t)n * P_DIM * Q_DIM;
    const float* Gb = G + (size_t)n * P_DIM * Q_DIM;

    v8f acc[2][4];
    const v8f vzero = {0.f,0.f,0.f,0.f,0.f,0.f,0.f,0.f};
#pragma unroll
    for (int t = 0; t < 2; ++t)
#pragma unroll
        for (int j = 0; j < 4; ++j) acc[t][j] = vzero;

    for (int c = 0; c < P_DIM / 64; ++c) {
        // stage 64x128 temp rows, k-pair interleaved (coalesced b64 global loads)
#pragma unroll
        for (int i = 0; i < 8; ++i) {
            int e  = tid + i * 256;          // 0..2047
            int pr = e >> 6;                 // k-pair row 0..31
            int cc = (e & 63) * 2;           // col 0..126 step 2
            size_t g0 = (size_t)(c * 64 + 2 * pr) * Q_DIM + cc;
            const float2 x0 = *(const float2*)(Xb + g0);
            const float2 x1 = *(const float2*)(Xb + g0 + Q_DIM);
            const float2 g2 = *(const float2*)(Gb + g0);
            const float2 g3 = *(const float2*)(Gb + g0 + Q_DIM);
            float4 st;                       // (k,c),(k+1,c),(k,c+1),(k+1,c+1)
            st.x = x0.x - lr * g2.x;
            st.y = x1.x - lr * g3.x;
            st.z = x0.y - lr * g2.y;
            st.w = x1.y - lr * g3.y;
            *(float4*)&lds[(pr * PITCH + cc) * 2] = st;
        }
        __syncthreads();
        for (int k = 0; k < 64; k += 4) {
            const int kk = k + 2 * half;
            const v2f a0 = pairfrag(lds, kk, m0 + lid);        // A = temp^T
            const v2f a1 = pairfrag(lds, kk, m0 + 16 + lid);
#pragma unroll
            for (int j = 0; j < 4; ++j) {
                const v2f b = pairfrag(lds, kk, n0 + j * 16 + lid); // B = temp
                acc[0][j] = wmma_f32(a0, b, acc[0][j]);
                acc[1][j] = wmma_f32(a1, b, acc[1][j]);
            }
        }
        __syncthreads();
    }
    float* Gm = gram + (size_t)n * Q_DIM * Q_DIM;
#pragma unroll
    for (int t = 0; t < 2; ++t)
#pragma unroll
        for (int j = 0; j < 4; ++j)
#pragma unroll
            for (int v = 0; v < 8; ++v) {
                int row = m0 + t * 16 + v + 8 * half;
                int col = n0 + j * 16 + lid;
                Gm[row * Q_DIM + col] = acc[t][j][v];
            }
}

// ---------------------------------------------------------------------------
// Kernel 2: Y = gram/||gram||_F (in place), Z = I, scale = ||gram||_F^{-1/2}
// ---------------------------------------------------------------------------
__global__ __launch_bounds__(256) void ns_prepare_kernel(const float* gram,
                                                         float* Y, float* Z,
                                                         float* scale) {
    __shared__ float red[256];
    const int n   = blockIdx.x;
    const int tid = threadIdx.x;
    const float* Gm = gram + (size_t)n * Q_DIM * Q_DIM;
    float s = 0.f;
    for (int i = tid; i < Q_DIM * Q_DIM; i += 256) { float v = Gm[i]; s += v * v; }
    red[tid] = s;
    __syncthreads();
    for (int off = 128; off > 0; off >>= 1) {
        if (tid < off) red[tid] += red[tid + off];
        __syncthreads();
    }
    const float fro = sqrtf(red[0]) + EPSV;
    const float inv = 1.0f / fro;
    if (tid == 0) scale[n] = rsqrtf(fro);
    float* Yb = Y + (size_t)n * Q_DIM * Q_DIM;
    float* Zb = Z + (size_t)n * Q_DIM * Q_DIM;
    for (int i = tid; i < Q_DIM * Q_DIM; i += 256) {
        Yb[i] = Gm[i] * inv;
        Zb[i] = ((i >> 7) == (i & 127)) ? 1.0f : 0.0f;
    }
}

// ---------------------------------------------------------------------------
// 128x128x128 matmul core: A row-major (b64 frags, pitch 36), B k-pair
// interleaved. Wave owns 2 row-blocks x 4 col-blocks.
// ---------------------------------------------------------------------------
__device__ __forceinline__ void matmul128(const float* A, const float* B,
                                          v8f acc[2][4], float* ldsA, float* ldsB,
                                          int tid, int wave, int lane) {
    const int half = lane >> 4;
    const int lid  = lane & 15;
    const int m0   = (wave & 3) * 32;
    const int n0   = (wave >> 2) * 64;
    const v8f vzero = {0.f,0.f,0.f,0.f,0.f,0.f,0.f,0.f};
#pragma unroll
    for (int t = 0; t < 2; ++t)
#pragma unroll
        for (int j = 0; j < 4; ++j) acc[t][j] = vzero;

    for (int c = 0; c < 4; ++c) {                      // K chunks of 32
#pragma unroll
        for (int i = 0; i < 4; ++i) {                  // A chunk: 128x32 row-major
            int e = tid + i * 256;
            int row = e >> 3;
            int col = (e & 7) * 4;
            *(float4*)&ldsA[row * 36 + col] =
                *(const float4*)(A + (size_t)row * 128 + c * 32 + col);
        }
#pragma unroll
        for (int i = 0; i < 4; ++i) {                  // B chunk: 32x128 interleaved
            int e  = tid + i * 256;                    // 0..1023
            int pr = e >> 6;                           // 0..15
            int cc = (e & 63) * 2;
            const float2 b0 = *(const float2*)(B + (size_t)(c * 32 + 2 * pr) * 128 + cc);
            const float2 b1 = *(const float2*)(B + (size_t)(c * 32 + 2 * pr + 1) * 128 + cc);
            float4 st;
            st.x = b0.x; st.y = b1.x; st.z = b0.y; st.w = b1.y;
            *(float4*)&ldsB[(pr * PITCH + cc) * 2] = st;
        }
        __syncthreads();
        for (int k = 0; k < 32; k += 4) {
            const int kk = k + 2 * half;
            const v2f a0 = *(const v2f*)&ldsA[(m0 + lid) * 36 + kk];
            const v2f a1 = *(const v2f*)&ldsA[(m0 + 16 + lid) * 36 + kk];
#pragma unroll
            for (int j = 0; j < 4; ++j) {
                const v2f b = pairfrag(ldsB, kk, n0 + j * 16 + lid);
                acc[0][j] = wmma_f32(a0, b, acc[0][j]);
                acc[1][j] = wmma_f32(a1, b, acc[1][j]);
            }
        }
        __syncthreads();
    }
}

// Kernel 3: T = 1.5*I - 0.5*(Z @ Y)
__global__ __launch_bounds__(256) void ns_T_kernel(const float* __restrict__ Z,
                                                   const float* __restrict__ Y,
                                                   float* __restrict__ T) {
    __shared__ __attribute__((aligned(16))) float ldsA[128 * 36];
    __shared__ __attribute__((aligned(16))) float ldsB[16 * PITCH * 2];
    const int n = blockIdx.x, tid = threadIdx.x;
    const int wave = tid >> 5, lane = tid & 31;
    const int half = lane >> 4, lid = lane & 15;
    const int m0 = (wave & 3) * 32, n0 = (wave >> 2) * 64;
    v8f acc[2][4];
    const size_t off = (size_t)n * Q_DIM * Q_DIM;
    matmul128(Z + off, Y + off, acc, ldsA, ldsB, tid, wave, lane);
    float* Tb = T + off;
#pragma unroll
    for (int t = 0; t < 2; ++t)
#pragma unroll
        for (int j = 0; j < 4; ++j)
#pragma unroll
            for (int v = 0; v < 8; ++v) {
                int row = m0 + t * 16 + v + 8 * half;
                int col = n0 + j * 16 + lid;
                Tb[row * Q_DIM + col] =
                    ((row == col) ? 1.5f : 0.0f) - 0.5f * acc[t][j][v];
            }
}

// Kernel 4: Y <- Y@T (y==0), Z <- T@Z (y==1). In-place safe per block.
__global__ __launch_bounds__(256) void ns_YZ_kernel(float* Y, float* Z,
                                                    const float* T) {
    __shared__ __attribute__((aligned(16))) float ldsA[128 * 36];
    __shared__ __attribute__((aligned(16))) float ldsB[16 * PITCH * 2];
    const int n = blockIdx.x, tid = threadIdx.x;
    const int wave = tid >> 5, lane = tid & 31;
    const int half = lane >> 4, lid = lane & 15;
    const int m0 = (wave & 3) * 32, n0 = (wave >> 2) * 64;
    const size_t off = (size_t)n * Q_DIM * Q_DIM;
    const float* A; const float* B; float* C;
    if (blockIdx.y == 0) { A = Y + off; B = T + off; C = Y + off; }
    else                 { A = T + off; B = Z + off; C = Z + off; }
    v8f acc[2][4];
    matmul128(A, B, acc, ldsA, ldsB, tid, wave, lane);
#pragma unroll
    for (int t = 0; t < 2; ++t)
#pragma unroll
        for (int j = 0; j < 4; ++j)
#pragma unroll
            for (int v = 0; v < 8; ++v)
                C[(m0 + t * 16 + v + 8 * half) * Q_DIM + n0 + j * 16 + lid] =
                    acc[t][j][v];
}

// ---------------------------------------------------------------------------
// Kernel 5: out = (X - lr*G) @ (Z * scale); grid.y = 128-row slab of P
// ---------------------------------------------------------------------------
__global__ __launch_bounds__(256) void final_kernel(const float* __restrict__ X,
                                                    const float* __restrict__ G,
                                                    const float* __restrict__ lr_p,
                                                    const float* __restrict__ Z,
                                                    const float* __restrict__ scale,
                                                    float* __restrict__ out) {
    __shared__ __attribute__((aligned(16))) float ldsA[128 * 36];
    __shared__ __attribute__((aligned(16))) float ldsB[16 * PITCH * 2];
    const int n  = blockIdx.x;
    const int p0 = blockIdx.y * 128;
    const float lr = lr_p[0];
    const int tid = threadIdx.x;
    const int wave = tid >> 5, lane = tid & 31;
    const int half = lane >> 4, lid = lane & 15;
    const int m0 = (wave & 3) * 32, n0 = (wave >> 2) * 64;

    const float* Xb = X + (size_t)n * P_DIM * Q_DIM + (size_t)p0 * Q_DIM;
    const float* Gb = G + (size_t)n * P_DIM * Q_DIM + (size_t)p0 * Q_DIM;
    const float* Zb = Z + (size_t)n * Q_DIM * Q_DIM;

    v8f acc[2][4];
    const v8f vzero = {0.f,0.f,0.f,0.f,0.f,0.f,0.f,0.f};
#pragma unroll
    for (int t = 0; t < 2; ++t)
#pragma unroll
        for (int j = 0; j < 4; ++j) acc[t][j] = vzero;

    for (int c = 0; c < 4; ++c) {
#pragma unroll
        for (int i = 0; i < 4; ++i) {                  // A = temp slab 128x32
            int e = tid + i * 256;
            int row = e >> 3;
            int col = (e & 7) * 4;
            size_t g = (size_t)row * Q_DIM + c * 32 + col;
            const float4 x4 = *(const float4*)(Xb + g);
            const float4 g4 = *(const float4*)(Gb + g);
            float4 t4;
            t4.x = x4.x - lr * g4.x;  t4.y = x4.y - lr * g4.y;
            t4.z = x4.z - lr * g4.z;  t4.w = x4.w - lr * g4.w;
            *(float4*)&ldsA[row * 36 + col] = t4;
        }
#pragma unroll
        for (int i = 0; i < 4; ++i) {                  // B = Z chunk, interleaved
            int e  = tid + i * 256;
            int pr = e >> 6;
            int cc = (e & 63) * 2;
            const float2 b0 = *(const float2*)(Zb + (size_t)(c * 32 + 2 * pr) * Q_DIM + cc);
            const float2 b1 = *(const float2*)(Zb + (size_t)(c * 32 + 2 * pr + 1) * Q_DIM + cc);
            float4 st;
            st.x = b0.x; st.y = b1.x; st.z = b0.y; st.w = b1.y;
            *(float4*)&ldsB[(pr * PITCH + cc) * 2] = st;
        }
        __syncthreads();
        for (int k = 0; k < 32; k += 4) {
            const int kk = k + 2 * half;
            const v2f a0 = *(const v2f*)&ldsA[(m0 + lid) * 36 + kk];
            const v2f a1 = *(const v2f*)&ldsA[(m0 + 16 + lid) * 36 + kk];
#pragma unroll
            for (int j = 0; j < 4; ++j) {
                const v2f b = pairfrag(ldsB, kk, n0 + j * 16 + lid);
                acc[0][j] = wmma_f32(a0, b, acc[0][j]);
                acc[1][j] = wmma_f32(a1, b, acc[1][j]);
            }
        }
        __syncthreads();
    }
    const float sc = scale[n];
    float* Ob = out + (size_t)n * P_DIM * Q_DIM + (size_t)p0 * Q_DIM;
#pragma unroll
    for (int t = 0; t < 2; ++t)
#pragma unroll
        for (int j = 0; j < 4; ++j)
#pragma unroll
            for (int v = 0; v < 8; ++v)
                Ob[(size_t)(m0 + t * 16 + v + 8 * half) * Q_DIM + n0 + j * 16 + lid] =
                    acc[t][j][v] * sc;
}

// ---------------------------------------------------------------------------
extern "C" void kernel_launch(void* const* d_in, const int* in_sizes, int n_in,
                              void* d_out, int out_size, void* d_ws, size_t ws_size,
                              hipStream_t stream) {
    (void)in_sizes; (void)n_in; (void)out_size; (void)ws_size;
    const float* X  = (const float*)d_in[0];
    const float* G  = (const float*)d_in[1];
    const float* lr = (const float*)d_in[2];
    float* out = (float*)d_out;

    const size_t MAT = (size_t)N_BATCH * Q_DIM * Q_DIM;
    float* Y     = (float*)d_ws;          // gram -> scaled in place
    float* Z     = Y + MAT;
    float* T     = Z + MAT;
    float* scale = T + MAT;

    dim3 blk(256);
    gram_kernel<<<N_BATCH, blk, 0, stream>>>(X, G, lr, Y);
    ns_prepare_kernel<<<N_BATCH, blk, 0, stream>>>(Y, Y, Z, scale);
    for (int it = 0; it < NS_ITERS; ++it) {
        ns_T_kernel<<<N_BATCH, blk, 0, stream>>>(Z, Y, T);
        ns_YZ_kernel<<<dim3(N_BATCH, 2), blk, 0, stream>>>(Y, Z, T);
    }
    final_kernel<<<dim3(N_BATCH, P_DIM / 128), blk, 0, stream>>>(X, G, lr, Z, scale, out);
}